// SingleTaskGIN_9612136808653
// MI455X (gfx1250) — compile-verified
//
#include <hip/hip_runtime.h>
#include <hip/hip_bf16.h>
#include <math.h>

#define N_NODES   100000
#define N_EDGES   1280000
#define NODE_DIM  128
#define HIDDEN    64
#define LAYERS    4
#define NUM_GRAPHS 512
#define BN_EPS    1e-5f

#define M_TILES   (N_NODES / 16)   // 6250, exact
#define WAVES_PER_BLOCK 8

typedef float v2f __attribute__((ext_vector_type(2)));
typedef float v8f __attribute__((ext_vector_type(8)));

// D = A(16x4 f32) * B(4x16 f32) + C, fp32 WMMA (wave32)
static __device__ __forceinline__ v8f wmma4(v2f a, v2f b, v8f c) {
  return __builtin_amdgcn_wmma_f32_16x16x4_f32(
      /*neg_a=*/false, a, /*neg_b=*/false, b,
      /*c_mod=*/(short)0, c, /*reuse_a=*/false, /*reuse_b=*/false);
}

// Async global->LDS copy of 16 bytes (per active lane). LDS byte address is
// the low 32 bits of the generic shared-aperture pointer (ISA 10.2).
// Tracked by ASYNCcnt; complete with s_wait_asynccnt before LDS reads.
static __device__ __forceinline__ void async_copy16(float* lds_dst,
                                                    const float* gsrc) {
  unsigned lds_off = (unsigned)(unsigned long long)lds_dst;
  asm volatile("global_load_async_to_lds_b128 %0, %1, off"
               :: "v"(lds_off), "v"(gsrc) : "memory");
}

static __device__ __forceinline__ void async_wait_and_barrier() {
  asm volatile("s_wait_asynccnt 0" ::: "memory");
  __syncthreads();
}

// ---------------------------------------------------------------------------
// h[N,64] = x[N,128] @ W.T(128,64) + b      (W stored [64,128] row-major)
// W staged in LDS via async copies; one wave per 16-row tile.
// ---------------------------------------------------------------------------
__global__ __launch_bounds__(256) void embed_kernel(
    const float* __restrict__ x, const float* __restrict__ W,
    const float* __restrict__ b, float* __restrict__ h) {
  __shared__ float wsh[HIDDEN * NODE_DIM];     // 8192 floats = 32 KB

  // cooperative async stage of the whole weight matrix (2048 x b128)
  const int tid = threadIdx.x;
#pragma unroll
  for (int i = 0; i < 8; ++i) {
    const int idx = (tid + i * 256) * 4;
    async_copy16(&wsh[idx], W + idx);
  }
  async_wait_and_barrier();

  const int wave = threadIdx.x >> 5;
  const int lane = threadIdx.x & 31;
  const int tile = blockIdx.x * WAVES_PER_BLOCK + wave;
  if (tile >= M_TILES) return;                 // wave-uniform: EXEC stays all-1s
  const int row0 = tile * 16;
  const int lr   = lane & 15;
  const int kh   = (lane >> 4) << 1;           // 0 for lanes 0-15, 2 for 16-31

  v8f acc[4] = {};
  const float* xrow = x + (size_t)(row0 + lr) * NODE_DIM;
  __builtin_prefetch(xrow, 0, 0);              // global_prefetch_b8
  __builtin_prefetch(xrow + 64, 0, 0);
#pragma unroll 4
  for (int k = 0; k < NODE_DIM; k += 4) {
    v2f a = *(const v2f*)(xrow + k + kh);      // A[M=lr][K=k+kh, k+kh+1]
#pragma unroll
    for (int t = 0; t < 4; ++t) {
      const int n = t * 16 + lr;
      v2f bv = *(const v2f*)(wsh + n * NODE_DIM + k + kh); // B[k][n]=W[n][k]
      acc[t] = wmma4(a, bv, acc[t]);
    }
  }
  const int rbase = row0 + ((lane >> 4) << 3); // lanes 16-31 hold M+8
#pragma unroll
  for (int t = 0; t < 4; ++t) {
    const int col = t * 16 + lr;
    const float bias = b[col];
#pragma unroll
    for (int r = 0; r < 8; ++r)
      h[(size_t)(rbase + r) * HIDDEN + col] = acc[t][r] + bias;
  }
}

// ---------------------------------------------------------------------------
// agg = h  (float4 copy; implements the GIN "+ self" term)
// ---------------------------------------------------------------------------
__global__ void copy4_kernel(const float4* __restrict__ src,
                             float4* __restrict__ dst, int n) {
  int i = blockIdx.x * blockDim.x + threadIdx.x;
  if (i < n) dst[i] = src[i];
}

// ---------------------------------------------------------------------------
// agg[dst[e]] += h[src[e]]   (edge-parallel, 16 float4 chunks per edge)
// ---------------------------------------------------------------------------
__global__ void edge_scatter_kernel(const int* __restrict__ src,
                                    const int* __restrict__ dst,
                                    const float* __restrict__ h,
                                    float* __restrict__ agg) {
  int tid = blockIdx.x * blockDim.x + threadIdx.x;
  if (tid >= N_EDGES * 16) return;
  const int e  = tid >> 4;
  const int c4 = (tid & 15) << 2;
  const int s = src[e], d = dst[e];
  float4 v = *(const float4*)(h + (size_t)s * HIDDEN + c4);
  float* ap = agg + (size_t)d * HIDDEN + c4;
  atomicAdd(ap + 0, v.x);
  atomicAdd(ap + 1, v.y);
  atomicAdd(ap + 2, v.z);
  atomicAdd(ap + 3, v.w);
}

__global__ void zero_kernel(float* __restrict__ p, int n) {
  int i = blockIdx.x * blockDim.x + threadIdx.x;
  if (i < n) p[i] = 0.f;
}

// ---------------------------------------------------------------------------
// m2 = relu(m @ W1.T + b1) @ W2.T + b2  (in place on m), + BN partial sums.
// W1/W2 staged in LDS via async copies; intermediate 16x64 tile transposed
// D-layout -> A-layout through a private per-wave LDS tile.
// ---------------------------------------------------------------------------
__global__ __launch_bounds__(256) void gin_mlp_kernel(
    float* __restrict__ m,
    const float* __restrict__ W1, const float* __restrict__ b1,
    const float* __restrict__ W2, const float* __restrict__ b2,
    float* __restrict__ bnsum, float* __restrict__ bnsumsq) {
  __shared__ float wsm[2 * HIDDEN * HIDDEN];           // W1 | W2, 32 KB
  __shared__ float tls[WAVES_PER_BLOCK * 16 * HIDDEN]; // staging, 32 KB

  const int tid = threadIdx.x;
#pragma unroll
  for (int i = 0; i < 4; ++i) {                        // 4096 floats of W1
    const int idx = (tid + i * 256) * 4;
    async_copy16(&wsm[idx], W1 + idx);
  }
#pragma unroll
  for (int i = 0; i < 4; ++i) {                        // 4096 floats of W2
    const int idx = (tid + i * 256) * 4;
    async_copy16(&wsm[HIDDEN * HIDDEN + idx], W2 + idx);
  }
  async_wait_and_barrier();

  const int wave = threadIdx.x >> 5;
  const int lane = threadIdx.x & 31;
  const int tile = blockIdx.x * WAVES_PER_BLOCK + wave;
  if (tile >= M_TILES) return;
  const int row0 = tile * 16;
  const int lr   = lane & 15;
  const int kh   = (lane >> 4) << 1;
  const float* w1s = wsm;
  const float* w2s = wsm + HIDDEN * HIDDEN;
  float* tl = tls + wave * 16 * HIDDEN;        // private per-wave staging tile

  // ---- GEMM 1: t = m @ W1.T ----
  v8f acc[4] = {};
  const float* arow = m + (size_t)(row0 + lr) * HIDDEN;
  __builtin_prefetch(arow, 0, 0);
#pragma unroll 4
  for (int k = 0; k < HIDDEN; k += 4) {
    v2f a = *(const v2f*)(arow + k + kh);
#pragma unroll
    for (int t = 0; t < 4; ++t) {
      v2f bv = *(const v2f*)(w1s + (t * 16 + lr) * HIDDEN + k + kh);
      acc[t] = wmma4(a, bv, acc[t]);
    }
  }
  // bias + ReLU, write to LDS (D-layout lanes scatter by column)
  const int rb = (lane >> 4) << 3;
#pragma unroll
  for (int t = 0; t < 4; ++t) {
    const int col = t * 16 + lr;
    const float bias = b1[col];
#pragma unroll
    for (int r = 0; r < 8; ++r) {
      float v = acc[t][r] + bias;
      tl[(rb + r) * HIDDEN + col] = v > 0.f ? v : 0.f;
    }
  }
  asm volatile("s_wait_dscnt 0" ::: "memory");  // LDS store -> transposed load

  // ---- GEMM 2: m2 = t @ W2.T ----
  v8f acc2[4] = {};
#pragma unroll 4
  for (int k = 0; k < HIDDEN; k += 4) {
    v2f a = *(const v2f*)(tl + lr * HIDDEN + k + kh);  // A[M=lr][K]
#pragma unroll
    for (int t = 0; t < 4; ++t) {
      v2f bv = *(const v2f*)(w2s + (t * 16 + lr) * HIDDEN + k + kh);
      acc2[t] = wmma4(a, bv, acc2[t]);
    }
  }
  // bias, store, BN partial sums (each lane owns one channel per t)
#pragma unroll
  for (int t = 0; t < 4; ++t) {
    const int col = t * 16 + lr;
    const float bias = b2[col];
    float s = 0.f, ss = 0.f;
#pragma unroll
    for (int r = 0; r < 8; ++r) {
      float v = acc2[t][r] + bias;
      m[(size_t)(row0 + rb + r) * HIDDEN + col] = v;
      s += v;
      ss += v * v;
    }
    atomicAdd(&bnsum[col], s);
    atomicAdd(&bnsumsq[col], ss);
  }
}

__global__ void bn_finalize_kernel(const float* __restrict__ bnsum,
                                   const float* __restrict__ bnsumsq,
                                   const float* __restrict__ gamma,
                                   const float* __restrict__ beta,
                                   float* __restrict__ scale,
                                   float* __restrict__ shift) {
  int c = threadIdx.x;
  if (c < HIDDEN) {
    float mean = bnsum[c] * (1.f / (float)N_NODES);
    float var  = bnsumsq[c] * (1.f / (float)N_NODES) - mean * mean;
    float sc   = gamma[c] * rsqrtf(var + BN_EPS);
    scale[c] = sc;
    shift[c] = beta[c] - mean * sc;
  }
}

__global__ void bn_apply_relu_kernel(const float* __restrict__ m,
                                     const float* __restrict__ scale,
                                     const float* __restrict__ shift,
                                     float* __restrict__ h) {
  int i = blockIdx.x * blockDim.x + threadIdx.x;
  if (i < N_NODES * HIDDEN) {
    int c = i & (HIDDEN - 1);
    float v = m[i] * scale[c] + shift[c];
    h[i] = v > 0.f ? v : 0.f;
  }
}

// ---------------------------------------------------------------------------
// g[batch[i]] += h[i]   (global_add_pool)
// ---------------------------------------------------------------------------
__global__ void pool_kernel(const float* __restrict__ h,
                            const int* __restrict__ batch,
                            float* __restrict__ g) {
  int tid = blockIdx.x * blockDim.x + threadIdx.x;
  if (tid >= N_NODES * 16) return;
  const int node = tid >> 4;
  const int c4   = (tid & 15) << 2;
  const int b = batch[node];
  float4 v = *(const float4*)(h + (size_t)node * HIDDEN + c4);
  float* gp = g + (size_t)b * HIDDEN + c4;
  atomicAdd(gp + 0, v.x);
  atomicAdd(gp + 1, v.y);
  atomicAdd(gp + 2, v.z);
  atomicAdd(gp + 3, v.w);
}

__global__ void fc1_kernel(const float* __restrict__ g,
                           const float* __restrict__ W,
                           const float* __restrict__ b,
                           float* __restrict__ g2) {
  int tid = blockIdx.x * blockDim.x + threadIdx.x;
  if (tid >= NUM_GRAPHS * HIDDEN) return;
  const int gr = tid >> 6, c = tid & 63;
  const float* gv = g + (size_t)gr * HIDDEN;
  const float* wv = W + (size_t)c * HIDDEN;
  float s = b[c];
#pragma unroll 8
  for (int k = 0; k < HIDDEN; ++k) s += gv[k] * wv[k];
  g2[tid] = s > 0.f ? s : 0.f;
}

__global__ void fc2_kernel(const float* __restrict__ g2,
                           const float* __restrict__ W,
                           const float* __restrict__ b,
                           float* __restrict__ out) {
  int gr = blockIdx.x * blockDim.x + threadIdx.x;
  if (gr >= NUM_GRAPHS) return;
  const float* gv = g2 + (size_t)gr * HIDDEN;
  float s = b[0];
#pragma unroll 8
  for (int k = 0; k < HIDDEN; ++k) s += gv[k] * W[k];
  out[gr] = s;
}

// ---------------------------------------------------------------------------
extern "C" void kernel_launch(void* const* d_in, const int* in_sizes, int n_in,
                              void* d_out, int out_size, void* d_ws, size_t ws_size,
                              hipStream_t stream) {
  const float* x       = (const float*)d_in[0];
  const int*   ei      = (const int*)  d_in[1];
  const int*   batch   = (const int*)  d_in[2];
  const float* embed_W = (const float*)d_in[3];
  const float* embed_b = (const float*)d_in[4];
  const float* conv_W1 = (const float*)d_in[5];
  const float* conv_b1 = (const float*)d_in[6];
  const float* conv_W2 = (const float*)d_in[7];
  const float* conv_b2 = (const float*)d_in[8];
  const float* bn_gamma= (const float*)d_in[9];
  const float* bn_beta = (const float*)d_in[10];
  const float* fc1_W   = (const float*)d_in[11];
  const float* fc1_b   = (const float*)d_in[12];
  const float* fc2_W   = (const float*)d_in[13];
  const float* fc2_b   = (const float*)d_in[14];
  const int* src = ei;
  const int* dst = ei + N_EDGES;

  // workspace layout (floats)
  float* ws      = (float*)d_ws;
  float* h       = ws;                                    // N*64
  float* agg     = h   + (size_t)N_NODES * HIDDEN;        // N*64 (m2 in place)
  float* bnsum   = agg + (size_t)N_NODES * HIDDEN;        // 64
  float* bnsumsq = bnsum + HIDDEN;                        // 64
  float* bnscale = bnsumsq + HIDDEN;                      // 64
  float* bnshift = bnscale + HIDDEN;                      // 64
  float* g       = bnshift + HIDDEN;                      // 512*64
  float* g2      = g + (size_t)NUM_GRAPHS * HIDDEN;       // 512*64

  const int gemm_blocks = (M_TILES + WAVES_PER_BLOCK - 1) / WAVES_PER_BLOCK; // 782

  embed_kernel<<<gemm_blocks, 256, 0, stream>>>(x, embed_W, embed_b, h);

  for (int l = 0; l < LAYERS; ++l) {
    // agg = h (self term), then scatter-add neighbors
    copy4_kernel<<<(N_NODES * 16 + 255) / 256, 256, 0, stream>>>(
        (const float4*)h, (float4*)agg, N_NODES * 16);
    edge_scatter_kernel<<<(N_EDGES * 16 + 255) / 256, 256, 0, stream>>>(
        src, dst, h, agg);
    // BN accumulators
    zero_kernel<<<1, 128, 0, stream>>>(bnsum, 2 * HIDDEN);
    // MLP (two WMMA GEMMs) + BN stats, in place on agg
    gin_mlp_kernel<<<gemm_blocks, 256, 0, stream>>>(
        agg,
        conv_W1 + (size_t)l * HIDDEN * HIDDEN, conv_b1 + (size_t)l * HIDDEN,
        conv_W2 + (size_t)l * HIDDEN * HIDDEN, conv_b2 + (size_t)l * HIDDEN,
        bnsum, bnsumsq);
    bn_finalize_kernel<<<1, 64, 0, stream>>>(
        bnsum, bnsumsq, bn_gamma + (size_t)l * HIDDEN, bn_beta + (size_t)l * HIDDEN,
        bnscale, bnshift);
    bn_apply_relu_kernel<<<(N_NODES * HIDDEN + 255) / 256, 256, 0, stream>>>(
        agg, bnscale, bnshift, h);
  }

  zero_kernel<<<(NUM_GRAPHS * HIDDEN + 255) / 256, 256, 0, stream>>>(
      g, NUM_GRAPHS * HIDDEN);
  pool_kernel<<<(N_NODES * 16 + 255) / 256, 256, 0, stream>>>(h, batch, g);
  fc1_kernel<<<(NUM_GRAPHS * HIDDEN + 255) / 256, 256, 0, stream>>>(
      g, fc1_W, fc1_b, g2);
  fc2_kernel<<<(NUM_GRAPHS + 255) / 256, 256, 0, stream>>>(
      g2, fc2_W, fc2_b, (float*)d_out);
}